// DGLGATModel_11081015623723
// MI455X (gfx1250) — compile-verified
//
#include <hip/hip_runtime.h>
#include <hip/hip_bf16.h>

#define NNODES 50000
#define NEDGES 800000
#define INF_   256
#define FDIM   32      // F1 == F2 == 32
#define HEADS  8
#define HF     (HEADS * FDIM)   // 256
#define NEG_SLOPE 0.2f

typedef float v2f __attribute__((ext_vector_type(2)));
typedef float v8f __attribute__((ext_vector_type(8)));

// ---------------- WMMA GEMM: Z[M,256] = X[M,K] * W[K,256], fp32 ----------------
// Block = 256 threads = 8 waves. Each wave computes TWO adjacent 16x16 column
// tiles (8 waves x 2 tiles = 16 tiles = all 256 columns), sharing the A loads
// and keeping two independent V_WMMA_F32_16X16X4_F32 accumulate chains in
// flight per wave. Grid = (M/16, 1).
__global__ __launch_bounds__(256) void gemm_wmma_f32(const float* __restrict__ X,
                                                     const float* __restrict__ W,
                                                     float* __restrict__ Z,
                                                     int K) {
    const int lane = threadIdx.x & 31;
    const int wave = threadIdx.x >> 5;
    const int half = lane >> 4;        // 0: lanes 0-15, 1: lanes 16-31
    const int mn   = lane & 15;
    const int row0 = blockIdx.x * 16;
    const int col0 = wave * 32;        // first tile
    const int col1 = col0 + 16;        // second tile

    const float* xrow = X + (size_t)(row0 + mn) * K;
    v8f c0 = {}, c1 = {};
    for (int k = 0; k < K; k += 4) {
        const int kk = k + half * 2;   // A lane L: row mn, K = kk, kk+1
        v2f a, b0, b1;
        a.x  = xrow[kk];
        a.y  = xrow[kk + 1];
        const float* w0 = W + (size_t)kk * HF;
        b0.x = w0[col0 + mn];
        b0.y = w0[HF + col0 + mn];
        b1.x = w0[col1 + mn];
        b1.y = w0[HF + col1 + mn];
        c0 = __builtin_amdgcn_wmma_f32_16x16x4_f32(false, a, false, b0,
                                                   (short)0, c0, false, false);
        c1 = __builtin_amdgcn_wmma_f32_16x16x4_f32(false, a, false, b1,
                                                   (short)0, c1, false, false);
    }
    // C/D layout: VGPR v -> row = row0 + v + half*8, col = colX + mn
    float* zout = Z + (size_t)(row0 + half * 8) * HF + mn;
#pragma unroll
    for (int v = 0; v < 8; ++v) {
        zout[(size_t)v * HF + col0] = c0[v];
        zout[(size_t)v * HF + col1] = c1[v];
    }
}

// ---------------- attention logits: el[n,h] = sum_f z[n,h,f]*al[h,f] ----------------
__global__ void attn_logits(const float* __restrict__ Z,
                            const float* __restrict__ al, const float* __restrict__ ar,
                            float* __restrict__ el, float* __restrict__ er) {
    int i = blockIdx.x * blockDim.x + threadIdx.x;   // i = n*H + h
    if (i >= NNODES * HEADS) return;
    const int h = i & (HEADS - 1);
    const int n = i >> 3;
    const float* z  = Z + (size_t)n * HF + h * FDIM;
    const float* pl = al + h * FDIM;
    const float* pr = ar + h * FDIM;
    float sl = 0.f, sr = 0.f;
#pragma unroll
    for (int f = 0; f < FDIM; ++f) { float zv = z[f]; sl += zv * pl[f]; sr += zv * pr[f]; }
    el[i] = sl; er[i] = sr;
}

// ---------------- order-preserving float <-> uint for atomicMax ----------------
__device__ __forceinline__ unsigned f2ord(float f) {
    unsigned u = __float_as_uint(f);
    return (u & 0x80000000u) ? ~u : (u | 0x80000000u);
}
__device__ __forceinline__ float ord2f(unsigned e) {
    unsigned u = (e & 0x80000000u) ? (e & 0x7FFFFFFFu) : ~e;
    return __uint_as_float(u);
}
#define ORD_NEG_INF 0x007FFFFFu   // f2ord(-inf)

__global__ void init_bufs(unsigned* __restrict__ m_ord, float* __restrict__ denom,
                          float* __restrict__ acc) {
    int i = blockIdx.x * blockDim.x + threadIdx.x;
    if (i < NNODES * HEADS) { m_ord[i] = ORD_NEG_INF; denom[i] = 0.f; }
    if (i < NNODES * FDIM)  acc[i] = 0.f;
}

__device__ __forceinline__ float edge_logit(const float* el, const float* er,
                                            int s, int d, int h) {
    float v = el[s * HEADS + h] + er[d * HEADS + h];
    return v >= 0.f ? v : NEG_SLOPE * v;     // LeakyReLU(0.2)
}

// pass 1: segmented max over incoming edges (per dst, per head)
__global__ void edge_max(const int* __restrict__ src, const int* __restrict__ dst,
                         const float* __restrict__ el, const float* __restrict__ er,
                         unsigned* __restrict__ m_ord) {
    int i = blockIdx.x * blockDim.x + threadIdx.x;   // i = e*H + h
    if (i >= NEDGES * HEADS) return;
    const int h = i & (HEADS - 1);
    const int e = i >> 3;
    const int s = src[e], d = dst[e];
    atomicMax(&m_ord[d * HEADS + h], f2ord(edge_logit(el, er, s, d, h)));
}

// pass 2: ex = exp(e - m[dst]); denom[dst] += ex
__global__ void edge_expsum(const int* __restrict__ src, const int* __restrict__ dst,
                            const float* __restrict__ el, const float* __restrict__ er,
                            const unsigned* __restrict__ m_ord,
                            float* __restrict__ exbuf, float* __restrict__ denom) {
    int i = blockIdx.x * blockDim.x + threadIdx.x;   // i = e*H + h
    if (i >= NEDGES * HEADS) return;
    const int h = i & (HEADS - 1);
    const int e = i >> 3;
    const int s = src[e], d = dst[e];
    float v  = edge_logit(el, er, s, d, h);
    float ex = __expf(v - ord2f(m_ord[d * HEADS + h]));
    exbuf[i] = ex;
    atomicAdd(&denom[d * HEADS + h], ex);
}

// pass 3: alpha[e,h] = ex/denom[dst] * 1/H, in-place in exbuf (one divide per (e,h))
__global__ void edge_alpha(const int* __restrict__ dst, const float* __restrict__ denom,
                           float* __restrict__ exbuf) {
    int i = blockIdx.x * blockDim.x + threadIdx.x;   // i = e*H + h
    if (i >= NEDGES * HEADS) return;
    const int h = i & (HEADS - 1);
    const int d = dst[i >> 3];
    exbuf[i] = exbuf[i] / denom[d * HEADS + h] * (1.0f / HEADS);
}

// pass 4: acc[dst,f] += sum_h alpha[e,h] * z[src,h,f]   (head-mean already folded)
__global__ void edge_agg(const int* __restrict__ src, const int* __restrict__ dst,
                         const float* __restrict__ Z, const float* __restrict__ exbuf,
                         float* __restrict__ acc) {
    int i = blockIdx.x * blockDim.x + threadIdx.x;   // i = e*F + f
    if (i >= NEDGES * FDIM) return;
    const int f = i & (FDIM - 1);
    const int e = i >> 5;
    const int s = src[e], d = dst[e];
    const float* zs = Z + (size_t)s * HF + f;
    const float* ae = exbuf + e * HEADS;
    float sum = 0.f;
#pragma unroll
    for (int h = 0; h < HEADS; ++h)
        sum += ae[h] * zs[h * FDIM];
    atomicAdd(&acc[d * FDIM + f], sum);
}

// epilogue: out = acc + mean_h(bias); optional ELU
__global__ void epilogue(const float* __restrict__ acc, const float* __restrict__ b,
                         float* __restrict__ out, int do_elu) {
    int i = blockIdx.x * blockDim.x + threadIdx.x;
    if (i >= NNODES * FDIM) return;
    const int f = i & (FDIM - 1);
    float bm = 0.f;
#pragma unroll
    for (int h = 0; h < HEADS; ++h) bm += b[h * FDIM + f];
    float v = acc[i] + bm * (1.0f / HEADS);
    out[i] = (do_elu && v <= 0.f) ? expm1f(v) : v;
}

// ---------------- host-side layer driver ----------------
static void run_gat_layer(const float* X, int K, const float* W,
                          const float* al, const float* ar, const float* b,
                          const int* src, const int* dst,
                          float* z, float* el, float* er, unsigned* m_ord,
                          float* denom, float* exbuf, float* acc, float* out,
                          int do_elu, hipStream_t stream) {
    gemm_wmma_f32<<<dim3(NNODES / 16, 1), 256, 0, stream>>>(X, W, z, K);

    int nh = NNODES * HEADS;
    attn_logits<<<(nh + 255) / 256, 256, 0, stream>>>(z, al, ar, el, er);

    int nf = NNODES * FDIM;
    init_bufs<<<(nf + 255) / 256, 256, 0, stream>>>(m_ord, denom, acc);

    int eh = NEDGES * HEADS;
    edge_max<<<(eh + 255) / 256, 256, 0, stream>>>(src, dst, el, er, m_ord);
    edge_expsum<<<(eh + 255) / 256, 256, 0, stream>>>(src, dst, el, er, m_ord, exbuf, denom);
    edge_alpha<<<(eh + 255) / 256, 256, 0, stream>>>(dst, denom, exbuf);

    int ef = NEDGES * FDIM;
    edge_agg<<<(ef + 255) / 256, 256, 0, stream>>>(src, dst, z, exbuf, acc);

    epilogue<<<(nf + 255) / 256, 256, 0, stream>>>(acc, b, out, do_elu);
}

extern "C" void kernel_launch(void* const* d_in, const int* in_sizes, int n_in,
                              void* d_out, int out_size, void* d_ws, size_t ws_size,
                              hipStream_t stream) {
    const float* x   = (const float*)d_in[0];
    const float* W1  = (const float*)d_in[1];
    const float* al1 = (const float*)d_in[2];
    const float* ar1 = (const float*)d_in[3];
    const float* b1  = (const float*)d_in[4];
    const float* W2  = (const float*)d_in[5];
    const float* al2 = (const float*)d_in[6];
    const float* ar2 = (const float*)d_in[7];
    const float* b2  = (const float*)d_in[8];
    const int*   src = (const int*)d_in[9];
    const int*   dst = (const int*)d_in[10];
    float* out = (float*)d_out;

    // workspace carve-out (floats/uints, all 4-byte)
    char* p = (char*)d_ws;
    float*    z     = (float*)p;    p += (size_t)NNODES * HF * 4;       // 51.2 MB
    float*    el    = (float*)p;    p += (size_t)NNODES * HEADS * 4;
    float*    er    = (float*)p;    p += (size_t)NNODES * HEADS * 4;
    unsigned* m_ord = (unsigned*)p; p += (size_t)NNODES * HEADS * 4;
    float*    denom = (float*)p;    p += (size_t)NNODES * HEADS * 4;
    float*    exbuf = (float*)p;    p += (size_t)NEDGES * HEADS * 4;    // 25.6 MB
    float*    acc   = (float*)p;    p += (size_t)NNODES * FDIM * 4;
    float*    hbuf  = (float*)p;    p += (size_t)NNODES * FDIM * 4;

    // layer 1: [N,256] -> GAT -> head-mean -> ELU -> hbuf [N,32]
    run_gat_layer(x, INF_, W1, al1, ar1, b1, src, dst,
                  z, el, er, m_ord, denom, exbuf, acc, hbuf, /*elu=*/1, stream);
    // layer 2: [N,32] -> GAT -> head-mean -> out [N,32]   (z buffer reused)
    run_gat_layer(hbuf, FDIM, W2, al2, ar2, b2, src, dst,
                  z, el, er, m_ord, denom, exbuf, acc, out, /*elu=*/0, stream);
}